// Global_model_19645180412553
// MI455X (gfx1250) — compile-verified
//
#include <hip/hip_runtime.h>
#include <math.h>

// ---------------------------------------------------------------------------
// MI455X (gfx1250) implementation.
// Convolutions are implicit GEMMs on V_WMMA_F32_16X16X4_F32 (fp32 matrix
// pipe, same precision as scalar FMA -> no accuracy risk vs the fp32
// reference). One wave32 owns a 16(pos) x 16(co) output tile; K = 9*Cin
// (zero-padded to a multiple of 4) is marched 4 at a time through the WMMA
// accumulator. A/B/D VGPR layouts per CDNA5 ISA 7.12.2:
//   A 16x4 : lane L supplies A[M=L&15][K = 2*(L>>4)+j] in vgpr j
//   B 4x16 : lane L supplies B[K = 2*(L>>4)+j][N=L&15] in vgpr j
//   D 16x16: vgpr g holds D[g + 8*(L>>4)][L&15]
// v2: branchless (clamped-address + select) input taps, weights pre-repacked
// into k-major order so the B fragment is one aligned b64 load per step.
// All non-conv stages are tiny (latency-bound) and use deterministic block
// reductions (no atomics, so graph replays are bit-stable).
// ---------------------------------------------------------------------------

typedef __attribute__((ext_vector_type(2))) float v2f;
typedef __attribute__((ext_vector_type(8))) float v8f;

__device__ __forceinline__ float blockReduceSum256(float v, float* sbuf) {
  const int tid = threadIdx.x;
  sbuf[tid] = v;
  __syncthreads();
#pragma unroll
  for (int s = 128; s > 0; s >>= 1) {
    if (tid < s) sbuf[tid] += sbuf[tid + s];
    __syncthreads();
  }
  float r = sbuf[0];
  __syncthreads();
  return r;
}

// ---------------- per-image mean of ref (y_exp) ----------------------------
__global__ __launch_bounds__(256) void k_mean_ref(const float* __restrict__ ref,
                                                  float* __restrict__ yexp,
                                                  int chw) {
  __shared__ float sbuf[256];
  const int n = blockIdx.x;
  const float* p = ref + (size_t)n * chw;
  float s = 0.f;
  for (int i = threadIdx.x; i < chw; i += 256) s += p[i];
  float S = blockReduceSum256(s, sbuf);
  if (threadIdx.x == 0) yexp[n] = S / (float)chw;
}

// ---------------- weight repack: (co,ci,3,3) -> (co, k) --------------------
// k = (kh*3+kw)*CINP + ci, zero padded for ci >= CIN.
template <int CIN, int COUT>
__global__ __launch_bounds__(256) void k_wrepack(const float* __restrict__ w,
                                                 float* __restrict__ wrep) {
  constexpr int CINP = (CIN + 3) & ~3;
  constexpr int KTOT = 9 * CINP;
  const int i = blockIdx.x * blockDim.x + threadIdx.x;
  if (i >= COUT * KTOT) return;
  const int co = i / KTOT;
  const int k = i - co * KTOT;
  const int s = k / CINP;
  const int ci = k - s * CINP;
  const int kh = s / 3;
  const int kw = s - kh * 3;
  wrep[i] = (ci < CIN) ? w[((co * CIN + ci) * 3 + kh) * 3 + kw] : 0.f;
}

// ---------------- conv3x3 stride2 pad1 + bias + LeakyReLU(0.2) -------------
// Implicit GEMM on V_WMMA_F32_16X16X4_F32.
template <int CIN, int COUT, int HIN, int WIN>
__global__ __launch_bounds__(256) void k_conv(const float* __restrict__ in,
                                              const float* __restrict__ wrep,
                                              const float* __restrict__ bias,
                                              float* __restrict__ out) {
  constexpr int HOUT = HIN / 2, WOUT = WIN / 2;
  constexpr int HWOUT = HOUT * WOUT;
  constexpr int CINP = (CIN + 3) & ~3;  // K padded to multiple of 4
  constexpr int KTOT = 9 * CINP;
  constexpr int COT = COUT / 16;

  const int lane = threadIdx.x & 31;
  const int wid = (blockIdx.x * blockDim.x + threadIdx.x) >> 5;
  const int posBase = (wid / COT) * 16;
  const int coBase = (wid % COT) * 16;

  const int m = lane & 15;           // A row / B col / D col for this lane
  const int koff = (lane >> 4) * 2;  // K sub-offset per ISA layout

  // address setup for this lane's A row (one output pixel)
  const int pos = posBase + m;
  const int n = pos / HWOUT;
  const int rem = pos - n * HWOUT;
  const int ph = rem / WOUT;
  const int pw = rem - ph * WOUT;
  const int hy0 = 2 * ph - 1;
  const int wx0 = 2 * pw - 1;
  const float* inN = in + (size_t)n * CIN * HIN * WIN;
  const float* wCol = wrep + (size_t)(coBase + m) * KTOT;  // B column, k-major

  v8f acc = {0.f, 0.f, 0.f, 0.f, 0.f, 0.f, 0.f, 0.f};
#pragma unroll 2
  for (int kb = 0; kb < KTOT; kb += 4) {
    // B fragment: pair (k, k+1) is contiguous & 8B aligned in repacked layout
    const v2f b = *(const v2f*)(wCol + kb + koff);
    v2f a;
#pragma unroll
    for (int j = 0; j < 2; ++j) {
      const int k = kb + koff + j;
      const int s = k / CINP;  // CINP is a power of two -> shift
      const int ci = k - s * CINP;
      const int kh = s / 3;
      const int kw = s - kh * 3;
      const int hy = hy0 + kh;
      const int wx = wx0 + kw;
      // branchless: clamp address into range, load unconditionally, select 0
      bool ok = (hy >= 0) && (hy < HIN) && (wx >= 0) && (wx < WIN);
      int cic = ci;
      if (CINP != CIN) {  // constexpr-foldable (only block 1)
        ok = ok && (ci < CIN);
        cic = (ci < CIN) ? ci : 0;
      }
      const int hyc = hy < 0 ? 0 : (hy >= HIN ? HIN - 1 : hy);
      const int wxc = wx < 0 ? 0 : (wx >= WIN ? WIN - 1 : wx);
      const float av = inN[(cic * HIN + hyc) * WIN + wxc];
      a[j] = ok ? av : 0.f;
    }
    acc = __builtin_amdgcn_wmma_f32_16x16x4_f32(false, a, false, b, (short)0,
                                                acc, false, false);
  }

  // write D tile with fused bias + LeakyReLU
  const int rowAdd = (lane >> 4) * 8;
  const int co = coBase + m;
  const float bco = bias[co];
#pragma unroll
  for (int g = 0; g < 8; ++g) {
    const int p2 = posBase + g + rowAdd;
    const int n2 = p2 / HWOUT;
    const int r2 = p2 - n2 * HWOUT;
    float v = acc[g] + bco;
    v = v >= 0.f ? v : 0.2f * v;
    out[((size_t)n2 * COUT + co) * HWOUT + r2] = v;
  }
}

// ---------------- FiLM coefficients: sc = softmax(y@W1+b1), sh likewise ----
__global__ void k_film(const float* __restrict__ y, int fcin,
                       const float* __restrict__ w1, const float* __restrict__ b1,
                       const float* __restrict__ w2, const float* __restrict__ b2,
                       float* __restrict__ sc, float* __restrict__ sh, int C) {
  __shared__ float sl[64];
  const int n = blockIdx.x, c = threadIdx.x;  // blockDim == C (<=64)
  const float* yp = y + (size_t)n * fcin;
  float l1 = b1[c], l2 = b2[c];
  for (int j = 0; j < fcin; ++j) {
    const float yv = yp[j];
    l1 = fmaf(yv, w1[j * C + c], l1);
    l2 = fmaf(yv, w2[j * C + c], l2);
  }
  // softmax over channels (stable)
  sl[c] = l1;
  __syncthreads();
  float mx = -INFINITY;
  for (int i = 0; i < C; ++i) mx = fmaxf(mx, sl[i]);
  __syncthreads();
  float e = expf(l1 - mx);
  sl[c] = e;
  __syncthreads();
  float sum = 0.f;
  for (int i = 0; i < C; ++i) sum += sl[i];
  sc[n * C + c] = e / sum;
  __syncthreads();
  sl[c] = l2;
  __syncthreads();
  mx = -INFINITY;
  for (int i = 0; i < C; ++i) mx = fmaxf(mx, sl[i]);
  __syncthreads();
  e = expf(l2 - mx);
  sl[c] = e;
  __syncthreads();
  sum = 0.f;
  for (int i = 0; i < C; ++i) sum += sl[i];
  sh[n * C + c] = e / sum;
}

// ---------------- instance norm (affine) + FiLM ----------------------------
__global__ __launch_bounds__(256) void k_instnorm(
    const float* __restrict__ x, float* __restrict__ y,
    const float* __restrict__ gw, const float* __restrict__ gb,
    const float* __restrict__ sc, const float* __restrict__ sh, int C, int HW) {
  __shared__ float sbuf[256];
  __shared__ float smean, sinv;
  const int nc = blockIdx.x;
  const int c = nc % C, n = nc / C;
  const float* xp = x + (size_t)nc * HW;
  float* yp = y + (size_t)nc * HW;
  float s = 0.f;
  for (int i = threadIdx.x; i < HW; i += 256) s += xp[i];
  float S = blockReduceSum256(s, sbuf);
  if (threadIdx.x == 0) smean = S / (float)HW;
  __syncthreads();
  const float mean = smean;
  float s2 = 0.f;
  for (int i = threadIdx.x; i < HW; i += 256) {
    const float d = xp[i] - mean;
    s2 += d * d;
  }
  float V = blockReduceSum256(s2, sbuf);
  if (threadIdx.x == 0) sinv = rsqrtf(V / (float)HW + 1e-5f);
  __syncthreads();
  const float scale = sinv * gw[c];
  const float fs = sc[n * C + c], fb = sh[n * C + c];
  const float a = scale * fs;
  const float b = (gb[c] - mean * scale) * fs + fb;
  for (int i = threadIdx.x; i < HW; i += 256) yp[i] = fmaf(xp[i], a, b);
}

// ---------------- heads ----------------------------------------------------
__global__ __launch_bounds__(256) void k_head_exp(const float* __restrict__ feat,
                                                  const float* __restrict__ w,
                                                  const float* __restrict__ b,
                                                  float* __restrict__ gamma) {
  __shared__ float sbuf[256];
  const int n = blockIdx.x;
  const float* fp = feat + (size_t)n * 4096;
  float s = 0.f;
  for (int j = threadIdx.x; j < 4096; j += 256) s = fmaf(fp[j], w[j], s);
  float S = blockReduceSum256(s, sbuf);
  if (threadIdx.x == 0) gamma[n] = tanhf(S + b[0]);
}

__global__ __launch_bounds__(256) void k_head_con(const float* __restrict__ feat,
                                                  const float* __restrict__ w,
                                                  const float* __restrict__ b,
                                                  float* __restrict__ cf) {
  __shared__ float sbuf[256];
  __shared__ float z[8];
  const int n = blockIdx.x;
  const float* fp = feat + (size_t)n * 4096;
  float acc[8] = {0.f, 0.f, 0.f, 0.f, 0.f, 0.f, 0.f, 0.f};
  for (int j = threadIdx.x; j < 4096; j += 256) {
    const float f = fp[j];
#pragma unroll
    for (int l = 0; l < 8; ++l) acc[l] = fmaf(f, w[j * 8 + l], acc[l]);
  }
  for (int l = 0; l < 8; ++l) {
    float S = blockReduceSum256(acc[l], sbuf);
    if (threadIdx.x == 0) z[l] = S + b[l];
  }
  __syncthreads();
  if (threadIdx.x == 0) {
    float mx = z[0];
    for (int l = 1; l < 8; ++l) mx = fmaxf(mx, z[l]);
    float e[8], sum = 0.f;
    for (int l = 0; l < 8; ++l) {
      e[l] = expf(z[l] - mx);
      sum += e[l];
    }
    for (int l = 0; l < 8; ++l) cf[n * 8 + l] = e[l] / sum;
  }
}

// ---------------- gamma curve + contrast step ------------------------------
__global__ __launch_bounds__(256) void k_gamma(const float* __restrict__ x,
                                               const float* __restrict__ gamma,
                                               float* __restrict__ xi, int chw) {
  __shared__ float sbuf[256];
  __shared__ float sth;
  const int n = blockIdx.x;
  const float g = gamma[n];
  const float* xp = x + (size_t)n * chw;
  float* op = xi + (size_t)n * chw;
  float s = 0.f;
  for (int i = threadIdx.x; i < chw; i += 256) {
    const float v = xp[i] + 1e-6f;
    float t = v / powf(v, g);
    t = fminf(fmaxf(t, 0.f), 1.f);
    op[i] = t;
    s += t;
  }
  float S = blockReduceSum256(s, sbuf);
  if (threadIdx.x == 0) sth = S / (float)chw;
  __syncthreads();
  const float th = sth;
  for (int i = threadIdx.x; i < chw; i += 256) {
    const float v = op[i];  // each thread re-reads only its own writes
    const float r = v + g * (v - th);
    op[i] = fminf(fmaxf(r, 0.f), 1.f);
  }
}

// ---------------- channel-max of ref, then soft histogram ------------------
__global__ __launch_bounds__(256) void k_refmax(const float* __restrict__ ref,
                                                float* __restrict__ mref) {
  const int i = blockIdx.x * blockDim.x + threadIdx.x;
  if (i >= 16 * 16384) return;
  const int n = i >> 14, p = i & 16383;
  const float* r = ref + (size_t)n * 3 * 16384;
  mref[i] = fmaxf(fmaxf(r[p], r[16384 + p]), r[2 * 16384 + p]);
}

// One block per image; thread tid owns bin tid (deterministic, no atomics).
__global__ __launch_bounds__(256) void k_hist(const float* __restrict__ mref,
                                              float* __restrict__ hist) {
  __shared__ float sbuf[256];
  __shared__ float sred[2];
  const int HW = 16384;
  const int n = blockIdx.x, tid = threadIdx.x;
  const float* mp = mref + (size_t)n * HW;
  float lmin = INFINITY, lmax = -INFINITY;
  for (int i = tid; i < HW; i += 256) {
    const float v = mp[i];
    lmin = fminf(lmin, v);
    lmax = fmaxf(lmax, v);
  }
  sbuf[tid] = lmin;
  __syncthreads();
  for (int s = 128; s > 0; s >>= 1) {
    if (tid < s) sbuf[tid] = fminf(sbuf[tid], sbuf[tid + s]);
    __syncthreads();
  }
  if (tid == 0) sred[0] = sbuf[0];
  __syncthreads();
  sbuf[tid] = lmax;
  __syncthreads();
  for (int s = 128; s > 0; s >>= 1) {
    if (tid < s) sbuf[tid] = fmaxf(sbuf[tid], sbuf[tid + s]);
    __syncthreads();
  }
  if (tid == 0) sred[1] = sbuf[0];
  __syncthreads();
  const float xmin = sred[0], xmax = sred[1];
  const float range = xmax - xmin;
  // mirror reference numerics: q_k and inter = q1 - q0
  const float qk = (2.0f * (float)tid + 1.0f) / 512.0f * range + xmin;
  const float q0 = (1.0f / 512.0f) * range + xmin;
  const float q1 = (3.0f / 512.0f) * range + xmin;
  const float inter = q1 - q0;
  const float thresh = 1.0f - inter;
  float sta = 0.f;
  for (int i = 0; i < HW; ++i) {  // uniform (broadcast) loads across the wave
    const float quant = 1.0f - fabsf(qk - mp[i]);
    if (quant > thresh) sta += quant;
  }
  const float S = blockReduceSum256(sta, sbuf);
  hist[n * 256 + tid] = sta / (S + 1e-6f);
}

// ---------------- piecewise-linear curve -----------------------------------
__global__ __launch_bounds__(256) void k_curve(const float* __restrict__ xi,
                                               const float* __restrict__ cf,
                                               float* __restrict__ out, int total,
                                               int chw) {
  const int i = blockIdx.x * blockDim.x + threadIdx.x;
  if (i >= total) return;
  const int n = i / chw;
  const float v = xi[i];
  float s = 0.f;
#pragma unroll
  for (int l = 0; l < 8; ++l) {
    const float t = fminf(fmaxf(8.f * v - (float)l, 0.f), 1.f);
    s = fmaf(t, cf[n * 8 + l], s);
  }
  out[i] = s;
}

__global__ void k_tail(const float* __restrict__ gam, const float* __restrict__ cf,
                       float* __restrict__ out) {
  const int i = threadIdx.x;
  if (i < 16) out[i * 9] = gam[i];
  if (i < 128) {
    const int n = i >> 3, l = i & 7;
    out[n * 9 + 1 + l] = cf[i];
  }
}

// ---------------------------------------------------------------------------
// Host orchestration
// ---------------------------------------------------------------------------
static void run_encoder(const float* input, const float* ycond, int fcin, int pbase,
                        void* const* d_in, float* bufA, float* bufB, float* sc,
                        float* sh, float* wrep, hipStream_t stream) {
  auto P = [d_in](int i) { return (const float*)d_in[i]; };
  const int N = 16;
  {  // block 1: 3 -> 16, 128 -> 64
    constexpr int CIN = 3, COUT = 16, HIN = 128, WIN = 128;
    constexpr int HOUT = HIN / 2, WOUT = WIN / 2;
    constexpr int KTOT = 9 * ((CIN + 3) & ~3);
    const int waves = (N * HOUT * WOUT / 16) * (COUT / 16);
    k_wrepack<CIN, COUT><<<(COUT * KTOT + 255) / 256, 256, 0, stream>>>(
        P(pbase + 0), wrep);
    k_conv<CIN, COUT, HIN, WIN><<<waves / 8, 256, 0, stream>>>(
        input, wrep, P(pbase + 1), bufA);
    k_film<<<N, COUT, 0, stream>>>(ycond, fcin, P(pbase + 4), P(pbase + 5),
                                   P(pbase + 6), P(pbase + 7), sc, sh, COUT);
    k_instnorm<<<N * COUT, 256, 0, stream>>>(bufA, bufB, P(pbase + 2),
                                             P(pbase + 3), sc, sh, COUT,
                                             HOUT * WOUT);
  }
  {  // block 2: 16 -> 32, 64 -> 32
    constexpr int CIN = 16, COUT = 32, HIN = 64, WIN = 64;
    constexpr int HOUT = HIN / 2, WOUT = WIN / 2;
    constexpr int KTOT = 9 * CIN;
    const int pb = pbase + 8;
    const int waves = (N * HOUT * WOUT / 16) * (COUT / 16);
    k_wrepack<CIN, COUT><<<(COUT * KTOT + 255) / 256, 256, 0, stream>>>(
        P(pb + 0), wrep);
    k_conv<CIN, COUT, HIN, WIN><<<waves / 8, 256, 0, stream>>>(
        bufB, wrep, P(pb + 1), bufA);
    k_film<<<N, COUT, 0, stream>>>(ycond, fcin, P(pb + 4), P(pb + 5), P(pb + 6),
                                   P(pb + 7), sc, sh, COUT);
    k_instnorm<<<N * COUT, 256, 0, stream>>>(bufA, bufB, P(pb + 2), P(pb + 3),
                                             sc, sh, COUT, HOUT * WOUT);
  }
  {  // block 3: 32 -> 64, 32 -> 16
    constexpr int CIN = 32, COUT = 64, HIN = 32, WIN = 32;
    constexpr int HOUT = HIN / 2, WOUT = WIN / 2;
    constexpr int KTOT = 9 * CIN;
    const int pb = pbase + 16;
    const int waves = (N * HOUT * WOUT / 16) * (COUT / 16);
    k_wrepack<CIN, COUT><<<(COUT * KTOT + 255) / 256, 256, 0, stream>>>(
        P(pb + 0), wrep);
    k_conv<CIN, COUT, HIN, WIN><<<waves / 8, 256, 0, stream>>>(
        bufB, wrep, P(pb + 1), bufA);
    k_film<<<N, COUT, 0, stream>>>(ycond, fcin, P(pb + 4), P(pb + 5), P(pb + 6),
                                   P(pb + 7), sc, sh, COUT);
    k_instnorm<<<N * COUT, 256, 0, stream>>>(bufA, bufB, P(pb + 2), P(pb + 3),
                                             sc, sh, COUT, HOUT * WOUT);
  }
  {  // block 4: 64 -> 64, 16 -> 8
    constexpr int CIN = 64, COUT = 64, HIN = 16, WIN = 16;
    constexpr int HOUT = HIN / 2, WOUT = WIN / 2;
    constexpr int KTOT = 9 * CIN;
    const int pb = pbase + 24;
    const int waves = (N * HOUT * WOUT / 16) * (COUT / 16);
    k_wrepack<CIN, COUT><<<(COUT * KTOT + 255) / 256, 256, 0, stream>>>(
        P(pb + 0), wrep);
    k_conv<CIN, COUT, HIN, WIN><<<waves / 8, 256, 0, stream>>>(
        bufB, wrep, P(pb + 1), bufA);
    k_film<<<N, COUT, 0, stream>>>(ycond, fcin, P(pb + 4), P(pb + 5), P(pb + 6),
                                   P(pb + 7), sc, sh, COUT);
    k_instnorm<<<N * COUT, 256, 0, stream>>>(bufA, bufB, P(pb + 2), P(pb + 3),
                                             sc, sh, COUT, HOUT * WOUT);
  }
}

extern "C" void kernel_launch(void* const* d_in, const int* in_sizes, int n_in,
                              void* d_out, int out_size, void* d_ws,
                              size_t ws_size, hipStream_t stream) {
  (void)in_sizes;
  (void)n_in;
  (void)out_size;
  (void)ws_size;
  // d_in layout (setup_inputs dict order, params flattened in insertion order):
  //   0: x (16,3,128,128)   1: ref (16,3,128,128)
  //   2..35  : "exp" params — per block i: conv_w, conv_b, in_w, in_b,
  //            fc1_w, fc1_b, fc2_w, fc2_b; then pre_w (4096,1), pre_b (1)
  //   36..69 : "con" params, same order; pre_w (4096,8), pre_b (8)
  const float* x = (const float*)d_in[0];
  const float* ref = (const float*)d_in[1];
  auto P = [d_in](int i) { return (const float*)d_in[i]; };

  float* ws = (float*)d_ws;
  float* bufA = ws;                  // 1,048,576 f  conv raw
  float* bufB = ws + 1048576;        // 1,048,576 f  normed activations
  float* xi = ws + 2097152;          //   786,432 f  gamma-adjusted image
  float* mref = ws + 2883584;        //   262,144 f  channel-max of ref
  float* yexp = ws + 3145728;        //        16 f
  float* gam = ws + 3145744;         //        16 f  gamma_factor
  float* hist = ws + 3145760;        //     4,096 f
  float* sc = ws + 3149856;          //     1,024 f
  float* sh = ws + 3150880;          //     1,024 f
  float* cf = ws + 3151904;          //       128 f
  float* wrep = ws + 3152032;        //    36,864 f  repacked conv weights
  const int CHW = 3 * 128 * 128;     // 49152

  // 1) exposure conditioning + "exp" encoder -> gamma
  k_mean_ref<<<16, 256, 0, stream>>>(ref, yexp, CHW);
  run_encoder(x, yexp, /*fcin=*/1, /*pbase=*/2, d_in, bufA, bufB, sc, sh, wrep,
              stream);
  k_head_exp<<<16, 256, 0, stream>>>(bufB, P(2 + 32), P(2 + 33), gam);

  // 2) gamma curve + contrast
  k_gamma<<<16, 256, 0, stream>>>(x, gam, xi, CHW);

  // 3) reference histogram
  k_refmax<<<(16 * 16384) / 256, 256, 0, stream>>>(ref, mref);
  k_hist<<<16, 256, 0, stream>>>(mref, hist);

  // 4) "con" encoder on xi -> curve coefficients
  run_encoder(xi, hist, /*fcin=*/256, /*pbase=*/36, d_in, bufA, bufB, sc, sh,
              wrep, stream);
  k_head_con<<<16, 256, 0, stream>>>(bufB, P(36 + 32), P(36 + 33), cf);

  // 5) apply piecewise-linear curve + emit (gamma, cf) tail
  k_curve<<<(16 * CHW) / 256, 256, 0, stream>>>(xi, cf, (float*)d_out, 16 * CHW,
                                                CHW);
  k_tail<<<1, 256, 0, stream>>>(gam, cf, (float*)d_out + 16 * CHW);
}